// AnQuantizer_59785944760420
// MI455X (gfx1250) — compile-verified
//
#include <hip/hip_runtime.h>
#include <stdint.h>

// ---------------------------------------------------------------------------
// A_n lattice quantizer, fused:  xp = x@T ; f = round/fix(xp) ; out = f@T^T
// FP32 WMMA (V_WMMA_F32_16X16X4_F32) — exactness required by round()+ranking.
// Guard-free inner loops + 1-deep software pipeline (loads for k+4 in flight
// while WMMA consumes k) so the scheduler is not forced to wait-to-zero.
// ---------------------------------------------------------------------------

typedef float v2f __attribute__((ext_vector_type(2)));
typedef float v8f __attribute__((ext_vector_type(8)));

#define DIM     256
#define N1      257            // DIM + 1
#define NPAD    260            // N1 padded up to multiple of 4 (K of GEMM2)
#define MTILE   16             // rows per block
#define THREADS 256            // 8 waves (wave32)
#define LSTRIDE 260            // LDS row stride in floats; 260 % 64 == 4 -> conflict-free

__device__ __forceinline__ uint64_t shflx_u64(uint64_t v, int m) {
  unsigned lo = (unsigned)v;
  unsigned hi = (unsigned)(v >> 32);
  lo = __shfl_xor(lo, m, 32);
  hi = __shfl_xor(hi, m, 32);
  return ((uint64_t)hi << 32) | lo;
}

__global__ __launch_bounds__(THREADS)
void an_quant_kernel(const float* __restrict__ x,
                     const float* __restrict__ T,   // [256][257] row-major
                     float* __restrict__ out,       // [batch][256]
                     int batch)
{
  __shared__ float xs[MTILE * LSTRIDE];  // x tile    (16,640 B)
  __shared__ float fp[MTILE * LSTRIDE];  // xp/f tile (16,640 B)

  const int tid     = threadIdx.x;
  const int lane    = tid & 31;
  const int wave    = tid >> 5;
  const int ln      = lane & 15;          // N-lane / M-lane within 16
  const int kh      = (lane >> 4) << 1;   // K sub-offset: 0 (lanes 0-15) or 2 (lanes 16-31)
  const int rowbase = blockIdx.x * MTILE;

  // ---------------- Phase 0: stage x tile into LDS (float4, coalesced) -----
  #pragma unroll
  for (int i = 0; i < 4; ++i) {
    int g   = tid + THREADS * i;          // 1024 float4 total, 4 per thread
    int row = g >> 6;
    int c4  = g & 63;
    float4 v = make_float4(0.f, 0.f, 0.f, 0.f);
    if (rowbase + row < batch)
      v = *(const float4*)&x[(size_t)(rowbase + row) * DIM + c4 * 4];
    *(float4*)&xs[row * LSTRIDE + c4 * 4] = v;
  }
  __syncthreads();

  // ---------------- Phase 1: xp = x @ T  (M=16, N=257, K=256) --------------
  // A 16x4 f32: lanes 0-15: M=lane, v={K0,K1}; lanes 16-31: M=lane-16, v={K2,K3}
  // B 4x16 f32: lanes 0-15: v={B[k0][n],B[k0+1][n]}; lanes 16-31: k0+=2
  // C 16x16 f32: c[r] -> M = r + 8*(lane>>4), N = lane&15
  //
  // Main tiles nt=0..15: columns always < 257 -> guard-free inner loop.
  for (int nt = wave; nt < 16; nt += 8) {
    const int n = nt * 16 + ln;                     // < 256, always valid
    const float* tb = T + n;                        // column n; element k at tb[k*N1]
    v8f acc = {0.f, 0.f, 0.f, 0.f, 0.f, 0.f, 0.f, 0.f};
    v2f a = *(const v2f*)&xs[ln * LSTRIDE + kh];
    v2f b;
    b.x = tb[(size_t)kh * N1];
    b.y = tb[(size_t)(kh + 1) * N1];
    #pragma unroll 4
    for (int kb = 4; kb < DIM; kb += 4) {
      const int k0 = kb + kh;
      v2f a2 = *(const v2f*)&xs[ln * LSTRIDE + k0];
      v2f b2;
      b2.x = tb[(size_t)k0 * N1];
      b2.y = tb[(size_t)(k0 + 1) * N1];
      acc = __builtin_amdgcn_wmma_f32_16x16x4_f32(
          false, a, false, b, (short)0, acc, false, false);
      a = a2; b = b2;
    }
    acc = __builtin_amdgcn_wmma_f32_16x16x4_f32(
        false, a, false, b, (short)0, acc, false, false);
    const int mrow = (lane >> 4) * 8;
    #pragma unroll
    for (int r = 0; r < 8; ++r)
      fp[(mrow + r) * LSTRIDE + nt * 16 + ln] = acc[r];
  }

  // Tail tile nt=16 (only column 256 is real): wave 0 only. Clamped address +
  // multiply mask -> unconditional loads, no exec-mask branching.
  if (wave == 0) {
    const int  n   = 256 + ln;
    const float mk = (n < N1) ? 1.f : 0.f;
    const float* tb = T + (n < N1 ? n : 0);
    v8f acc = {0.f, 0.f, 0.f, 0.f, 0.f, 0.f, 0.f, 0.f};
    #pragma unroll 4
    for (int kb = 0; kb < DIM; kb += 4) {
      const int k0 = kb + kh;
      v2f b;
      b.x = tb[(size_t)k0 * N1] * mk;
      b.y = tb[(size_t)(k0 + 1) * N1] * mk;
      v2f a = *(const v2f*)&xs[ln * LSTRIDE + k0];
      acc = __builtin_amdgcn_wmma_f32_16x16x4_f32(
          false, a, false, b, (short)0, acc, false, false);
    }
    const int mrow = (lane >> 4) * 8;
    #pragma unroll
    for (int r = 0; r < 8; ++r) {
      const int col = 256 + ln;
      if (col < NPAD) fp[(mrow + r) * LSTRIDE + col] = acc[r]; // cols 257..259 = exact 0
    }
  }
  __syncthreads();

  // ---------------- Phase 2: round + zero-sum correction (1 wave / row) ----
  for (int row = wave; row < MTILE; row += 8) {
    float* fr = &fp[row * LSTRIDE];
    uint64_t kv[9];              // (monotone f32 key << 32) | index
    float    part = 0.f;
    #pragma unroll
    for (int j = 0; j < 9; ++j) {
      const int idx = lane + 32 * j;
      if (idx < N1) {
        const float v  = fr[idx];
        const float fv = __builtin_rintf(v);       // RNE == jnp.round
        const float d  = v - fv;                   // residual in [-0.5, 0.5]
        fr[idx] = fv;
        part   += fv;                              // small exact integers
        const uint32_t u   = __float_as_uint(d);
        const uint32_t key = (u & 0x80000000u) ? ~u : (u | 0x80000000u);
        kv[j] = ((uint64_t)key << 32) | (uint32_t)idx;
      } else {
        kv[j] = 0;  // fixed to direction-specific sentinel below
      }
    }
    for (int off = 16; off; off >>= 1) part += __shfl_xor(part, off, 32);
    const int Delta = (int)part;

    if (Delta > 0) {
      // subtract 1 from the Delta smallest residuals (ties: smallest index)
      #pragma unroll
      for (int j = 0; j < 9; ++j) if (lane + 32 * j >= N1) kv[j] = ~0ull;
      for (int it = 0; it < Delta; ++it) {
        uint64_t m = kv[0];
        #pragma unroll
        for (int j = 1; j < 9; ++j) m = (kv[j] < m) ? kv[j] : m;
        uint64_t g = m;
        for (int off = 16; off; off >>= 1) {
          const uint64_t o = shflx_u64(g, off);
          g = (o < g) ? o : g;
        }
        if (m == g) {                              // unique owner (keys distinct)
          #pragma unroll
          for (int j = 0; j < 9; ++j)
            if (kv[j] == g) { kv[j] = ~0ull; fr[(uint32_t)g] -= 1.f; }
        }
      }
    } else if (Delta < 0) {
      // add 1 to the |Delta| largest residuals (ties: largest index)
      for (int it = 0; it < -Delta; ++it) {
        uint64_t m = kv[0];
        #pragma unroll
        for (int j = 1; j < 9; ++j) m = (kv[j] > m) ? kv[j] : m;
        uint64_t g = m;
        for (int off = 16; off; off >>= 1) {
          const uint64_t o = shflx_u64(g, off);
          g = (o > g) ? o : g;
        }
        if (m == g) {
          #pragma unroll
          for (int j = 0; j < 9; ++j)
            if (kv[j] == g) { kv[j] = 0; fr[(uint32_t)g] += 1.f; }
        }
      }
    }
  }
  __syncthreads();

  // ---------------- Phase 3: out = f @ T^T  (M=16, N=256, K=257) -----------
  // B[k][n] = T[n][k]. Main loop covers k=0..255 guard-free; k=256 peeled.
  for (int nt = wave; nt < 16; nt += 8) {
    const int n = nt * 16 + ln;                    // output column, < 256
    const float* tr = T + (size_t)n * N1;          // row n of T; element k at tr[k]
    v8f acc = {0.f, 0.f, 0.f, 0.f, 0.f, 0.f, 0.f, 0.f};
    v2f a = *(const v2f*)&fp[ln * LSTRIDE + kh];
    v2f b;
    b.x = tr[kh];
    b.y = tr[kh + 1];
    #pragma unroll 4
    for (int kb = 4; kb < DIM; kb += 4) {          // k0 max = 255 -> guard-free
      const int k0 = kb + kh;
      v2f a2 = *(const v2f*)&fp[ln * LSTRIDE + k0];
      v2f b2;
      b2.x = tr[k0];
      b2.y = tr[k0 + 1];
      acc = __builtin_amdgcn_wmma_f32_16x16x4_f32(
          false, a, false, b, (short)0, acc, false, false);
      a = a2; b = b2;
    }
    acc = __builtin_amdgcn_wmma_f32_16x16x4_f32(
        false, a, false, b, (short)0, acc, false, false);
    // peeled k-step: only K=256 contributes (fp cols 257..259 are exact zeros)
    {
      const float t256 = tr[256];
      v2f bt;
      bt.x = (kh == 0) ? t256 : 0.f;
      bt.y = 0.f;
      v2f at = *(const v2f*)&fp[ln * LSTRIDE + 256 + kh];
      acc = __builtin_amdgcn_wmma_f32_16x16x4_f32(
          false, at, false, bt, (short)0, acc, false, false);
    }
    const int mrow = (lane >> 4) * 8;
    #pragma unroll
    for (int r = 0; r < 8; ++r) {
      const int grow = rowbase + mrow + r;
      if (grow < batch)
        out[(size_t)grow * DIM + nt * 16 + ln] = acc[r];  // coalesced f32 stores
    }
  }
}

extern "C" void kernel_launch(void* const* d_in, const int* in_sizes, int n_in,
                              void* d_out, int out_size, void* d_ws, size_t ws_size,
                              hipStream_t stream) {
  (void)n_in; (void)out_size; (void)d_ws; (void)ws_size;
  const float* x   = (const float*)d_in[0];
  const float* T   = (const float*)d_in[1];   // [256][257] float32
  float*       out = (float*)d_out;
  const int batch  = in_sizes[0] / DIM;
  const int grid   = (batch + MTILE - 1) / MTILE;
  an_quant_kernel<<<grid, THREADS, 0, stream>>>(x, T, out, batch);
}